// PNA_model2_26843545600700
// MI455X (gfx1250) — compile-verified
//
#include <hip/hip_runtime.h>
#include <hip/hip_bf16.h>
#include <math.h>

#define N_NODES  30000
#define N_EDGES  150000
#define N_GRAPHS 64
#define TOWERS   5
#define F_IN     75
#define F_OUT    15
#define HSTRIDE  76            // padded feature stride, h[..][75] == 0
#define MSTRIDE  375           // TOWERS * F_IN
#define KPAD     152           // 76 + 76 (per-half pad), multiple of 4
#define APITCH   156           // LDS row pitch for A tile (bank-friendly)
#define NTILES_E (N_EDGES/16)  // 9375 edge tiles

typedef __attribute__((ext_vector_type(2))) float v2f;
typedef __attribute__((ext_vector_type(8))) float v8f;

// ---------------------------------------------------------------- utilities
__global__ void k_zero_i32(int* p, int n) {
    int i = blockIdx.x * blockDim.x + threadIdx.x;
    if (i < n) p[i] = 0;
}

__global__ void k_hist(const int* __restrict__ dst, int* __restrict__ deg) {
    int e = blockIdx.x * blockDim.x + threadIdx.x;
    if (e < N_EDGES) atomicAdd(&deg[dst[e]], 1);
}

// exclusive scan of deg[30000] -> row_ptr[30001]; single block of 1024
__global__ __launch_bounds__(1024) void k_scan(const int* __restrict__ deg,
                                               int* __restrict__ rp) {
    __shared__ int part[1024];
    const int tid = threadIdx.x;
    const int CH = 30;                       // 1024*30 = 30720 >= 30000
    int base = tid * CH;
    int loc[CH];
    int s = 0;
    for (int i = 0; i < CH; ++i) {
        int idx = base + i;
        int v = (idx < N_NODES) ? deg[idx] : 0;
        loc[i] = s; s += v;
    }
    part[tid] = s;
    __syncthreads();
    for (int off = 1; off < 1024; off <<= 1) {
        int v = 0;
        if (tid >= off) v = part[tid - off];
        __syncthreads();
        if (tid >= off) part[tid] += v;
        __syncthreads();
    }
    int excl = (tid == 0) ? 0 : part[tid - 1];
    for (int i = 0; i < CH; ++i) {
        int idx = base + i;
        if (idx < N_NODES) rp[idx] = excl + loc[i];
    }
    if (tid == 1023) rp[N_NODES] = part[1023];
}

__global__ void k_cursor(const int* __restrict__ rp, int* __restrict__ cur) {
    int i = blockIdx.x * blockDim.x + threadIdx.x;
    if (i < N_NODES) cur[i] = rp[i];
}

__global__ void k_scatter(const int* __restrict__ src, const int* __restrict__ dst,
                          int* __restrict__ cur, int* __restrict__ geid,
                          int* __restrict__ gsrc, int* __restrict__ gdst) {
    int e = blockIdx.x * blockDim.x + threadIdx.x;
    if (e >= N_EDGES) return;
    int d = dst[e];
    int pos = atomicAdd(&cur[d], 1);
    geid[pos] = e;
    gsrc[pos] = src[e];
    gdst[pos] = d;
}

// deterministic canonical order: per-node insertion sort by edge id
__global__ void k_sortseg(const int* __restrict__ rp, int* __restrict__ geid,
                          int* __restrict__ gsrc) {
    int n = blockIdx.x * blockDim.x + threadIdx.x;
    if (n >= N_NODES) return;
    int s = rp[n], e = rp[n + 1];
    for (int i = s + 1; i < e; ++i) {
        int ke = geid[i], ks = gsrc[i];
        int j = i - 1;
        while (j >= s && geid[j] > ke) {
            geid[j + 1] = geid[j]; gsrc[j + 1] = gsrc[j]; --j;
        }
        geid[j + 1] = ke; gsrc[j + 1] = ks;
    }
}

// --------------------------------------------------------------- pre-layer
__global__ void k_pre(const float* __restrict__ x, const float* __restrict__ pw,
                      const float* __restrict__ pb, float* __restrict__ h,
                      float* __restrict__ out_fixed) {
    int idx = blockIdx.x * blockDim.x + threadIdx.x;
    if (idx >= N_NODES * HSTRIDE) return;
    int n = idx / HSTRIDE, j = idx % HSTRIDE;
    if (j < F_IN) {
        h[idx] = x[n * 2 + 0] * pw[j] + x[n * 2 + 1] * pw[F_IN + j] + pb[j];
        if (j == 0) out_fixed[n] = x[n * 2 + 0];
    } else {
        h[idx] = 0.f;
    }
}

// --------------------------------------------- edge message GEMM via WMMA f32
// grid: (XB, 25)  y -> (tower, ntile); block: 128 threads (4 waves)
__global__ __launch_bounds__(128) void k_edge_gemm(
    const float* __restrict__ h, const float* __restrict__ Wp_l,
    const float* __restrict__ bp_l, const int* __restrict__ gsrc,
    const int* __restrict__ gdst, float* __restrict__ m) {

    const int t = blockIdx.y / 5;
    const int ntile = blockIdx.y % 5;
    const int colbase = ntile * 16;
    const float* W = Wp_l + t * (2 * F_IN) * F_IN;

    __shared__ float Bs[KPAD * 16];
    __shared__ float As[4][16 * APITCH];

    // stage B column strip: Bs[k][c] = W[wrow(k)][colbase+c] (0 padded)
    for (int i = threadIdx.x; i < KPAD * 16; i += 128) {
        int k = i >> 4, c = i & 15;
        int col = colbase + c;
        float v = 0.f;
        int wr = -1;
        if (k < 76) { if (k < F_IN) wr = k; }
        else        { int kk = k - 76; if (kk < F_IN) wr = F_IN + kk; }
        if (wr >= 0 && col < F_IN) v = W[wr * F_IN + col];
        Bs[i] = v;
    }
    __syncthreads();

    const int wave = threadIdx.x >> 5;
    const int lane = threadIdx.x & 31;
    const int r  = lane & 15;      // row (A) / col (B,D)
    const int hs = lane >> 4;      // K half-select
    float* A = &As[wave][0];

    for (int tile = blockIdx.x * 4 + wave; tile < NTILES_E; tile += gridDim.x * 4) {
        const int ebase = tile * 16;
        // gather A tile: 16 edges x 152 feats ([h[dst](76) | h[src](76)])
        for (int i = lane; i < 16 * KPAD; i += 32) {
            int row = i / KPAD, col = i % KPAD;
            int node = (col < 76) ? gdst[ebase + row] : gsrc[ebase + row];
            int f = (col < 76) ? col : col - 76;      // f==75 -> padded 0 in h
            A[row * APITCH + col] = h[node * HSTRIDE + f];
        }
        asm volatile("s_wait_dscnt 0" ::: "memory");  // LDS writes visible to wave

        v8f acc = {};
        for (int ks = 0; ks < KPAD; ks += 4) {
            v2f a, b;
            a.x = A[r * APITCH + ks + 2 * hs + 0];
            a.y = A[r * APITCH + ks + 2 * hs + 1];
            b.x = Bs[(ks + 2 * hs + 0) * 16 + r];
            b.y = Bs[(ks + 2 * hs + 1) * 16 + r];
            acc = __builtin_amdgcn_wmma_f32_16x16x4_f32(
                false, a, false, b, (short)0, acc, false, false);
        }

        const int col = colbase + r;
        if (col < F_IN) {
            const float bb = bp_l[t * F_IN + col];
#pragma unroll
            for (int v = 0; v < 8; ++v) {
                int row = v + 8 * hs;
                m[(size_t)(ebase + row) * MSTRIDE + t * F_IN + col] = acc[v] + bb;
            }
        }
        asm volatile("" ::: "memory");
    }
}

// ------------------------------------- per-node aggregation + post + W_lin
__global__ __launch_bounds__(384) void k_agg_post(
    const float* __restrict__ h, const float* __restrict__ m,
    const int* __restrict__ rp, const float* __restrict__ avgp,
    const float* __restrict__ Wq_l, const float* __restrict__ bq_l,
    const float* __restrict__ Wl_l, const float* __restrict__ bl_l,
    float* __restrict__ hout) {

    const int n = blockIdx.x;
    const int tid = threadIdx.x;
    __shared__ float aggL[TOWERS * 300];
    __shared__ float hL[HSTRIDE];
    __shared__ float outL[TOWERS * F_OUT];

    const int start = rp[n], end = rp[n + 1];
    const float dg = (float)(end - start);

    if (tid < HSTRIDE) hL[tid] = h[n * HSTRIDE + tid];

    if (tid < MSTRIDE) {
        float s = 0.f, s2 = 0.f, mn = INFINITY, mx = -INFINITY;
        for (int e = start; e < end; ++e) {
            float v = m[(size_t)e * MSTRIDE + tid];
            s += v; s2 += v * v;
            mn = fminf(mn, v); mx = fmaxf(mx, v);
        }
        float cnt = fmaxf(dg, 1.f);
        float mean = s / cnt, mean2 = s2 / cnt;
        float var = mean2 - mean * mean; if (var < 0.f) var = 0.f;
        float sd = sqrtf(var + 1e-5f);
        if (!(dg > 0.f)) { mn = 0.f; mx = 0.f; }
        int t = tid / F_IN, j = tid % F_IN;
        aggL[t * 300 + j]        = mean;
        aggL[t * 300 + 75 + j]   = mn;
        aggL[t * 300 + 150 + j]  = mx;
        aggL[t * 300 + 225 + j]  = sd;
    }
    __syncthreads();

    const float amp  = log1pf(fmaxf(dg, 1.f)) / avgp[0];
    const float iamp = 1.f / amp;

    if (tid < TOWERS * F_OUT) {                       // 75 threads
        int t = tid / F_OUT, o = tid % F_OUT;
        const float* W = Wq_l + t * 975 * F_OUT;
        float acc = bq_l[t * F_OUT + o];
        for (int j = 0; j < F_IN; ++j) acc += hL[j] * W[j * F_OUT + o];
        const float* ag = &aggL[t * 300];
        for (int j = 0; j < 300; ++j) {
            float a = ag[j];
            acc += a * (W[(75 + j) * F_OUT + o]
                        + amp  * W[(375 + j) * F_OUT + o]
                        + iamp * W[(675 + j) * F_OUT + o]);
        }
        outL[tid] = acc;
    }
    __syncthreads();

    if (tid < HSTRIDE) {
        if (tid < F_IN) {
            float acc = bl_l[tid];
            for (int k = 0; k < F_IN; ++k) acc += outL[k] * Wl_l[k * F_IN + tid];
            hout[n * HSTRIDE + tid] = acc;
        } else {
            hout[n * HSTRIDE + F_IN] = 0.f;
        }
    }
}

// ------------------------------------------------- batchnorm (deterministic)
__global__ __launch_bounds__(256) void k_bnstats(const float* __restrict__ h,
                                                 float* __restrict__ stats) {
    const int j = blockIdx.x;                         // feature column, 75
    __shared__ double ssum[256], ssq[256];
    double s = 0.0, q = 0.0;
    for (int n = threadIdx.x; n < N_NODES; n += 256) {
        double v = (double)h[n * HSTRIDE + j];
        s += v; q += v * v;
    }
    ssum[threadIdx.x] = s; ssq[threadIdx.x] = q;
    __syncthreads();
    for (int o = 128; o > 0; o >>= 1) {
        if (threadIdx.x < o) {
            ssum[threadIdx.x] += ssum[threadIdx.x + o];
            ssq[threadIdx.x]  += ssq[threadIdx.x + o];
        }
        __syncthreads();
    }
    if (threadIdx.x == 0) {
        double mu = ssum[0] / (double)N_NODES;
        double var = ssq[0] / (double)N_NODES - mu * mu;
        stats[j] = (float)mu;
        stats[F_IN + j] = (float)var;
    }
}

__global__ void k_bnrelu(float* __restrict__ h, const float* __restrict__ stats,
                         const float* __restrict__ g, const float* __restrict__ b) {
    int idx = blockIdx.x * blockDim.x + threadIdx.x;
    if (idx >= N_NODES * HSTRIDE) return;
    int j = idx % HSTRIDE;
    if (j < F_IN) {
        float v = h[idx];
        v = g[j] * (v - stats[j]) * rsqrtf(stats[F_IN + j] + 1e-5f) + b[j];
        h[idx] = fmaxf(v, 0.f);
    } else {
        h[idx] = 0.f;
    }
}

// --------------------------------------------------------------- final MLP
__global__ __launch_bounds__(64) void k_mlp(
    const float* __restrict__ h, const float* __restrict__ alpha,
    const int* __restrict__ batch, const float* __restrict__ noise,
    const float* __restrict__ mw1, const float* __restrict__ mb1,
    const float* __restrict__ mw2, const float* __restrict__ mb2,
    const float* __restrict__ mw3, const float* __restrict__ mb3,
    float* __restrict__ out) {

    const int n = blockIdx.x;
    const int tid = threadIdx.x;
    __shared__ float hb[76], a1[50], a2[25];

    for (int i = tid; i < F_IN; i += 64) hb[i] = h[n * HSTRIDE + i];
    if (tid == 0) hb[75] = alpha[batch[n]];
    __syncthreads();

    for (int j = tid; j < 50; j += 64) {
        float acc = mb1[j];
        for (int i = 0; i < 76; ++i) acc += hb[i] * mw1[i * 50 + j];
        a1[j] = fmaxf(acc, 0.f);
    }
    __syncthreads();

    if (tid < 25) {
        float acc = mb2[tid];
        for (int i = 0; i < 50; ++i) acc += a1[i] * mw2[i * 25 + tid];
        a2[tid] = fmaxf(acc, 0.f);
    }
    __syncthreads();

    if (tid == 0) {
        float acc = mb3[0];
        for (int i = 0; i < 25; ++i) acc += a2[i] * mw3[i];
        float v = acc + noise[n];
        out[n] = 1.f / (1.f + expf(-v));
    }
}

// ---------------------------------------------------------------- launcher
extern "C" void kernel_launch(void* const* d_in, const int* in_sizes, int n_in,
                              void* d_out, int out_size, void* d_ws, size_t ws_size,
                              hipStream_t stream) {
    const float* x        = (const float*)d_in[0];
    const float* alpha    = (const float*)d_in[1];
    const int*   eidx     = (const int*)  d_in[2];
    const int*   batch    = (const int*)  d_in[3];
    const float* noise    = (const float*)d_in[4];
    const float* avgp     = (const float*)d_in[5];
    const float* pre_w    = (const float*)d_in[6];
    const float* pre_b    = (const float*)d_in[7];
    const float* W_pre    = (const float*)d_in[8];
    const float* b_pre    = (const float*)d_in[9];
    const float* W_post   = (const float*)d_in[10];
    const float* b_post   = (const float*)d_in[11];
    const float* W_lin    = (const float*)d_in[12];
    const float* b_lin    = (const float*)d_in[13];
    const float* bn_g     = (const float*)d_in[14];
    const float* bn_b     = (const float*)d_in[15];
    const float* mw1      = (const float*)d_in[16];
    const float* mb1      = (const float*)d_in[17];
    const float* mw2      = (const float*)d_in[18];
    const float* mb2      = (const float*)d_in[19];
    const float* mw3      = (const float*)d_in[20];
    const float* mb3      = (const float*)d_in[21];

    const int* src = eidx;
    const int* dst = eidx + N_EDGES;

    // workspace carve (256B aligned)
    char* p = (char*)d_ws;
    auto carve = [&](size_t bytes) -> void* {
        void* r = (void*)p;
        p += (bytes + 255) & ~(size_t)255;
        return r;
    };
    float* h_a   = (float*)carve((size_t)N_NODES * HSTRIDE * 4);
    float* h_b   = (float*)carve((size_t)N_NODES * HSTRIDE * 4);
    float* m     = (float*)carve((size_t)N_EDGES * MSTRIDE * 4);
    int*   deg   = (int*)  carve((size_t)N_NODES * 4);
    int*   rp    = (int*)  carve((size_t)(N_NODES + 1) * 4);
    int*   cur   = (int*)  carve((size_t)N_NODES * 4);
    int*   geid  = (int*)  carve((size_t)N_EDGES * 4);
    int*   gsrc  = (int*)  carve((size_t)N_EDGES * 4);
    int*   gdst  = (int*)  carve((size_t)N_EDGES * 4);
    float* stats = (float*)carve(160 * 4);

    float* out_main  = (float*)d_out;            // sigmoid outputs [0, 30000)
    float* out_fixed = out_main + N_NODES;       // fixed feature   [30000, 60000)

    // CSR build
    k_zero_i32<<<(N_NODES + 255) / 256, 256, 0, stream>>>(deg, N_NODES);
    k_hist<<<(N_EDGES + 255) / 256, 256, 0, stream>>>(dst, deg);
    k_scan<<<1, 1024, 0, stream>>>(deg, rp);
    k_cursor<<<(N_NODES + 255) / 256, 256, 0, stream>>>(rp, cur);
    k_scatter<<<(N_EDGES + 255) / 256, 256, 0, stream>>>(src, dst, cur, geid, gsrc, gdst);
    k_sortseg<<<(N_NODES + 255) / 256, 256, 0, stream>>>(rp, geid, gsrc);

    // pre-linear + fixed feature
    k_pre<<<(N_NODES * HSTRIDE + 255) / 256, 256, 0, stream>>>(x, pre_w, pre_b, h_a, out_fixed);

    float* hcur = h_a;
    float* hnext = h_b;
    for (int l = 0; l < 4; ++l) {
        const float* Wp_l = W_pre  + (size_t)l * TOWERS * 2 * F_IN * F_IN;
        const float* bp_l = b_pre  + (size_t)l * TOWERS * F_IN;
        const float* Wq_l = W_post + (size_t)l * TOWERS * 975 * F_OUT;
        const float* bq_l = b_post + (size_t)l * TOWERS * F_OUT;
        const float* Wl_l = W_lin  + (size_t)l * F_IN * F_IN;
        const float* bl_l = b_lin  + (size_t)l * F_IN;

        k_edge_gemm<<<dim3(128, 25), 128, 0, stream>>>(hcur, Wp_l, bp_l, gsrc, gdst, m);
        k_agg_post<<<N_NODES, 384, 0, stream>>>(hcur, m, rp, avgp, Wq_l, bq_l, Wl_l, bl_l, hnext);
        k_bnstats<<<F_IN, 256, 0, stream>>>(hnext, stats);
        k_bnrelu<<<(N_NODES * HSTRIDE + 255) / 256, 256, 0, stream>>>(
            hnext, stats, bn_g + l * F_IN, bn_b + l * F_IN);

        float* tmp = hcur; hcur = hnext; hnext = tmp;
    }

    k_mlp<<<N_NODES, 64, 0, stream>>>(hcur, alpha, batch, noise,
                                      mw1, mb1, mw2, mb2, mw3, mb3, out_main);
}